// StudentFPMaskGenerator_88115549045403
// MI455X (gfx1250) — compile-verified
//
#include <hip/hip_runtime.h>
#include <hip/hip_bf16.h>
#include <math.h>

// CDNA5 / gfx1250 wave32. WMMA f32 16x16x4 computes the pairwise squared-
// distance matrix for (16 pixels) x (16 boxes) per instruction.

typedef __attribute__((ext_vector_type(2))) float v2f;
typedef __attribute__((ext_vector_type(8))) float v8f;

#define FH 128
#define FW 128
#define NBOX 1024
#define NBATCH 4
#define SCALE 0.8f          // VOXEL * OUT_FACTOR
#define PC_MIN (-51.2f)
#define OVERLAP 0.2f
#define MIN_RADIUS 2
#define EPSF 1.1920928955078125e-07f   // jnp.finfo(float32).eps
#define MAXS 1040                       // 1024 survivors + pad to /16
#define TPB 512                         // 16 waves; 1 wave per tile row

__device__ __forceinline__ float gauss_radius(float h, float w) {
  const float ov = OVERLAP;
  float b1 = h + w;
  float c1 = w * h * (1.0f - ov) / (1.0f + ov);
  float sq1 = sqrtf(fmaxf(b1 * b1 - 4.0f * c1, 0.0f));
  float r1 = 0.5f * (b1 + sq1);

  float b2 = 2.0f * (h + w);
  float c2 = (1.0f - ov) * w * h;
  float sq2 = sqrtf(fmaxf(b2 * b2 - 16.0f * c2, 0.0f));
  float r2 = 0.5f * (b2 + sq2);

  float a3 = 4.0f * ov;
  float b3 = -2.0f * ov * (h + w);
  float c3 = (ov - 1.0f) * w * h;
  float sq3 = sqrtf(fmaxf(b3 * b3 - 4.0f * a3 * c3, 0.0f));
  float r3 = 0.5f * (b3 + sq3);
  return fminf(fminf(r1, r2), r3);
}

__global__ __launch_bounds__(TPB)
void StudentFPMaskGenerator_88115549045403_kernel(
    const float* __restrict__ boxes,   // (B, N, 7): x y z w l h yaw
    const float* __restrict__ scores,  // (B, N)
    float* __restrict__ out) {         // (B, 1, FH, FW)
  __shared__ int   s_cnt;
  __shared__ int   s_cx[MAXS];
  __shared__ int   s_cy[MAXS];
  __shared__ int   s_rad[MAXS];
  __shared__ float s_i2s[MAXS];   // 1 / (2*sigma^2)
  __shared__ float s_sc[MAXS];

  const int tid = threadIdx.x;
  const int b   = blockIdx.z;
  const int tx0 = blockIdx.x * 16;
  const int ty0 = blockIdx.y * 16;

  if (tid == 0) s_cnt = 0;
  __syncthreads();

  // ---- Stage 1: per-tile box cull + param precompute into LDS ----
  for (int n = tid; n < NBOX; n += TPB) {
    const float* bx = boxes + ((size_t)b * NBOX + n) * 7;
    float x = bx[0], y = bx[1], w = bx[3], l = bx[4];
    float wf = w / SCALE;
    float lf = l / SCALE;
    bool valid = (wf > 0.0f) && (lf > 0.0f) && (wf <= 1000.0f) && (lf <= 1000.0f);
    float r = valid ? gauss_radius(lf, wf) : (float)MIN_RADIUS;
    int rad = max(MIN_RADIUS, (int)floorf(r));
    int cx = (int)((x - PC_MIN) / SCALE);   // trunc-toward-zero, matches astype(int32)
    int cy = (int)((y - PC_MIN) / SCALE);
    valid = valid && (cx >= 0) && (cx < FW) && (cy >= 0) && (cy < FH);
    bool hit = valid && (cx + rad >= tx0) && (cx - rad <= tx0 + 15)
                     && (cy + rad >= ty0) && (cy - rad <= ty0 + 15);
    if (hit) {
      int idx = atomicAdd(&s_cnt, 1);
      float sg = (2.0f * (float)rad + 1.0f) * (1.0f / 6.0f);
      s_cx[idx]  = cx;
      s_cy[idx]  = cy;
      s_rad[idx] = rad;
      s_i2s[idx] = 0.5f / (sg * sg);
      s_sc[idx]  = scores[(size_t)b * NBOX + n];
    }
  }
  __syncthreads();
  const int cnt    = s_cnt;
  const int padded = (cnt + 15) & ~15;
  for (int i = cnt + tid; i < padded; i += TPB) {  // sentinel pad: contributes 0
    s_cx[i] = -10000; s_cy[i] = -10000; s_rad[i] = -1;
    s_i2s[i] = 0.0f;  s_sc[i] = 0.0f;
  }
  __syncthreads();

  // ---- Stage 2: WMMA distance matrix + exp/window/max epilogue ----
  const int lane = tid & 31;
  const int row  = tid >> 5;        // wave id -> tile row 0..15
  const int hi   = lane >> 4;       // lane half selects K{0,1} vs K{2,3} / M+8
  const int nlo  = lane & 15;
  const int py   = ty0 + row;
  const float pyf = (float)py;
  const float pxf = (float)(tx0 + nlo);   // x of pixel M = nlo

  // A (16 pixels x K=4): row M = [px, py, px^2+py^2, 1]
  v2f afrag;
  if (hi == 0) { afrag.x = pxf;                     afrag.y = pyf;  }
  else         { afrag.x = pxf * pxf + pyf * pyf;   afrag.y = 1.0f; }

  float m[8];
#pragma unroll
  for (int v = 0; v < 8; ++v) m[v] = 0.0f;

  for (int cb = 0; cb < padded; cb += 16) {
    const int j  = cb + nlo;        // this lane's box column
    const int cx = s_cx[j], cy = s_cy[j], rad = s_rad[j];
    const float i2s = s_i2s[j], sc = s_sc[j];
    const float cxf = (float)cx, cyf = (float)cy;

    // B (K=4 x 16 boxes): col N = [-2cx, -2cy, 1, cx^2+cy^2]
    v2f bfrag;
    if (hi == 0) { bfrag.x = -2.0f * cxf;  bfrag.y = -2.0f * cyf;          }
    else         { bfrag.x = 1.0f;         bfrag.y = cxf * cxf + cyf * cyf; }

    v8f acc = {0.f, 0.f, 0.f, 0.f, 0.f, 0.f, 0.f, 0.f};
    // d2[p,n] = (px-cx)^2 + (py-cy)^2, exact in f32 (all integer operands)
    v8f d2 = __builtin_amdgcn_wmma_f32_16x16x4_f32(
        false, afrag, false, bfrag, (short)0, acc, false, false);

    const int dy = py - cy;
    const bool oky = (dy <= rad) && (-dy <= rad);   // rad = -1 sentinel fails
#pragma unroll
    for (int v = 0; v < 8; ++v) {
      // C/D layout: VGPR v, lane half -> pixel M = v + 8*hi, box N = nlo
      const int dx = tx0 + v + 8 * hi - cx;
      const float g = __expf(-d2[v] * i2s);
      const bool okw = oky && (dx <= rad) && (-dx <= rad) && (g >= EPSF);
      m[v] = fmaxf(m[v], okw ? g * sc : 0.0f);
    }
  }

  // max-reduce the 16 box columns within each lane half (pixels M=0..7 / 8..15)
#pragma unroll
  for (int s = 1; s < 16; s <<= 1) {
#pragma unroll
    for (int v = 0; v < 8; ++v)
      m[v] = fmaxf(m[v], __shfl_xor(m[v], s, 32));
  }

  if (nlo == 0) {   // lanes 0 and 16 each store 8 contiguous pixels of this row
    size_t base = ((size_t)b * FH + py) * FW + (size_t)(tx0 + 8 * hi);
#pragma unroll
    for (int v = 0; v < 8; ++v) out[base + v] = m[v];
  }
}

extern "C" void kernel_launch(void* const* d_in, const int* in_sizes, int n_in,
                              void* d_out, int out_size, void* d_ws, size_t ws_size,
                              hipStream_t stream) {
  (void)in_sizes; (void)n_in; (void)d_ws; (void)ws_size; (void)out_size;
  const float* boxes  = (const float*)d_in[0];
  const float* scores = (const float*)d_in[1];
  float* out = (float*)d_out;
  dim3 grid(FW / 16, FH / 16, NBATCH);   // 8 x 8 tiles x 4 batches = 256 blocks
  StudentFPMaskGenerator_88115549045403_kernel<<<grid, TPB, 0, stream>>>(boxes, scores, out);
}